// GuidedFilter_71287867179172
// MI455X (gfx1250) — compile-verified
//
#include <hip/hip_runtime.h>
#include <stdint.h>

// Guided filter, r=4, eps=1e-8, N=4, C_y=1, C_x=3, H=W=512, fp32.
// One wave32 per 16x16 tile. LDS tiles filled by the Tensor Data Mover;
// horizontal box sums done as V(16x24) x B(24x16) via v_wmma_f32_16x16x4_f32.

#define RR    4
#define TSZ   16
#define HTW   (TSZ + 2 * RR)   // 24
#define IMW   512
#define IMH   512
#define NBATCH 4
#define EPSF  1e-8f

typedef unsigned int uint32x4_t __attribute__((ext_vector_type(4)));
typedef int          int32x8_t  __attribute__((ext_vector_type(8)));
typedef int          int32x4_t  __attribute__((ext_vector_type(4)));
typedef float        v2f        __attribute__((ext_vector_type(2)));
typedef float        v8f        __attribute__((ext_vector_type(8)));

__device__ __forceinline__ int rfl_i(int v) { return __builtin_amdgcn_readfirstlane(v); }
__device__ __forceinline__ unsigned rfl_u(unsigned v) {
  return (unsigned)__builtin_amdgcn_readfirstlane((int)v);
}

__device__ __forceinline__ uint32_t lds_byte_off(const void* p) {
  return (uint32_t)(uintptr_t)p;   // low 32 bits of generic LDS pointer = LDS byte offset
}

__device__ __forceinline__ int iabs(int v) { return v < 0 ? -v : v; }

// One 2D TDM tile load: Wc x Hc fp32 from a row-major plane (pitch strideElems)
// into packed LDS (pitch = Wc floats) at lds_addr. D# per CDNA5 ISA ch.8.
__device__ __forceinline__ void tdm_load_tile_f32(uint32_t lds_addr, const float* gsrc,
                                                  int Wc, int Hc, int strideElems) {
  uint64_t ga = (uint64_t)(uintptr_t)gsrc;

  uint32x4_t g0 = {
      rfl_u(1u),                                                  // count=1 (valid)
      rfl_u(lds_addr),                                            // lds_addr
      rfl_u((uint32_t)ga),                                        // global_addr[31:0]
      rfl_u(((uint32_t)(ga >> 32) & 0x1FFFFFFu) | (2u << 30))     // addr[56:32] | type=2
  };

  uint32_t td0 = (uint32_t)strideElems;   // tensor_dim0
  uint32_t td1 = (uint32_t)IMH;           // tensor_dim1
  uint64_t st0 = (uint64_t)strideElems;   // tensor_dim0_stride (elements)

  int32x8_t g1 = {
      rfl_i(2 << 16),                                             // data_size=2 (4B)
      rfl_i((int)((td0 & 0xFFFFu) << 16)),                        // tensor_dim0 lo
      rfl_i((int)((td0 >> 16) | ((td1 & 0xFFFFu) << 16))),        // dim0 hi | dim1 lo
      rfl_i((int)((td1 >> 16) | ((uint32_t)Wc << 16))),           // dim1 hi | tile_dim0
      rfl_i((int)(uint32_t)Hc),                                   // tile_dim1 (tile_dim2=0)
      rfl_i((int)(uint32_t)(st0 & 0xFFFFFFFFull)),                // dim0_stride lo
      rfl_i((int)(uint32_t)((st0 >> 32) & 0xFFFFull)),            // dim0_stride hi
      rfl_i(0)
  };
  int32x4_t z4 = {0, 0, 0, 0};
  int32x8_t z8 = {0, 0, 0, 0, 0, 0, 0, 0};
  __builtin_amdgcn_tensor_load_to_lds(g0, g1, z4, z4, z8, 0);
}

// ---------------- Stage 1: per-pixel A(3), b(1) ----------------
__global__ __launch_bounds__(32)
void gf_stage1(const float* __restrict__ y, const float* __restrict__ x,
               float* __restrict__ ab) {
  __shared__ float raw[4][HTW * HTW];   // y, x0, x1, x2 (packed, pitch = Wc)
  __shared__ float vs[13][TSZ * HTW];   // vertical sums, fixed pitch 24, zero-padded

  const int n    = blockIdx.z;
  const int h0   = blockIdx.y * TSZ;
  const int w0   = blockIdx.x * TSZ;
  const int lane = threadIdx.x;

  const int r0 = max(h0 - RR, 0);
  const int r1 = min(h0 + TSZ - 1 + RR, IMH - 1);
  const int c0 = max(w0 - RR, 0);
  const int c1 = min(w0 + TSZ - 1 + RR, IMW - 1);
  const int Hc = r1 - r0 + 1;
  const int Wc = c1 - c0 + 1;

  const size_t HW = (size_t)IMH * IMW;

  // --- TDM fill of 4 raw planes (wave-level, uniform descriptors) ---
  {
    const float* py = y + (size_t)n * HW + (size_t)r0 * IMW + c0;
    tdm_load_tile_f32(lds_byte_off(&raw[0][0]), py, Wc, Hc, IMW);
    for (int c = 0; c < 3; ++c) {
      const float* px = x + ((size_t)n * 3 + c) * HW + (size_t)r0 * IMW + c0;
      tdm_load_tile_f32(lds_byte_off(&raw[1 + c][0]), px, Wc, Hc, IMW);
    }
    __builtin_amdgcn_s_wait_tensorcnt(0);
  }
  __syncthreads();

  // --- Vertical clipped 9-tap sums of the 13 product channels ---
  for (int idx = lane; idx < TSZ * HTW; idx += 32) {
    const int orow = idx / HTW;
    const int cc   = idx - orow * HTW;
    float s0=0,s1=0,s2=0,s3=0,s4=0,s5=0,s6=0,s7=0,s8=0,s9=0,s10=0,s11=0,s12=0;
    if (cc < Wc) {
      const int h   = h0 + orow;
      const int rlo = max(h - RR, 0) - r0;
      const int rhi = min(h + RR, IMH - 1) - r0;
      for (int rr = rlo; rr <= rhi; ++rr) {
        const int off = rr * Wc + cc;
        const float vy = raw[0][off];
        const float v0 = raw[1][off];
        const float v1 = raw[2][off];
        const float v2 = raw[3][off];
        s0 += v0;      s1 += v1;      s2 += v2;      s3 += vy;
        s4 += vy * v0; s5 += vy * v1; s6 += vy * v2;
        s7 += v0 * v0; s8 += v0 * v1; s9 += v0 * v2;
        s10 += v1 * v1; s11 += v1 * v2; s12 += v2 * v2;
      }
    }
    vs[0][idx]=s0;  vs[1][idx]=s1;  vs[2][idx]=s2;  vs[3][idx]=s3;
    vs[4][idx]=s4;  vs[5][idx]=s5;  vs[6][idx]=s6;
    vs[7][idx]=s7;  vs[8][idx]=s8;  vs[9][idx]=s9;
    vs[10][idx]=s10; vs[11][idx]=s11; vs[12][idx]=s12;
  }
  __syncthreads();

  // --- Horizontal sums as T = V(16x24) x B(24x16) via WMMA f32 16x16x4 ---
  const int d0    = w0 - c0;            // 4 interior, 0 at left border
  const int Nn    = lane & 15;          // B/D column, also A row (M)
  const int khalf = (lane >> 4) * 2;    // K pair selected by half-wave

  v8f acc[13];
  #pragma unroll
  for (int chn = 0; chn < 13; ++chn) acc[chn] = (v8f){0, 0, 0, 0, 0, 0, 0, 0};

  #pragma unroll
  for (int kb = 0; kb < 6; ++kb) {
    const int k0 = kb * 4 + khalf;
    const v2f bfrag = { (iabs(k0     - Nn - d0) <= RR) ? 1.0f : 0.0f,
                        (iabs(k0 + 1 - Nn - d0) <= RR) ? 1.0f : 0.0f };
    #pragma unroll
    for (int chn = 0; chn < 13; ++chn) {
      const v2f afrag = { vs[chn][Nn * HTW + k0], vs[chn][Nn * HTW + k0 + 1] };
      acc[chn] = __builtin_amdgcn_wmma_f32_16x16x4_f32(
          false, afrag, false, bfrag, (short)0, acc[chn], false, false);
    }
  }

  // --- Per-lane: 8 pixels (D layout: VGPR vi -> M = vi + 8*(lane>>4), N = lane&15) ---
  const int hi = lane >> 4;
  const int w  = w0 + Nn;
  const int nhc = min(w + RR, IMW - 1) - max(w - RR, 0) + 1;

  #pragma unroll
  for (int vi = 0; vi < 8; ++vi) {
    const int M = vi + 8 * hi;
    const int h = h0 + M;
    const int nvc = min(h + RR, IMH - 1) - max(h - RR, 0) + 1;
    const float invN = 1.0f / (float)(nvc * nhc);

    const float mx0 = acc[0][vi] * invN, mx1 = acc[1][vi] * invN, mx2 = acc[2][vi] * invN;
    const float my  = acc[3][vi] * invN;
    const float cy0 = acc[4][vi] * invN - my * mx0;
    const float cy1 = acc[5][vi] * invN - my * mx1;
    const float cy2 = acc[6][vi] * invN - my * mx2;
    const float c00 = acc[7][vi]  * invN - mx0 * mx0 + EPSF;
    const float c01 = acc[8][vi]  * invN - mx0 * mx1;
    const float c02 = acc[9][vi]  * invN - mx0 * mx2;
    const float c11 = acc[10][vi] * invN - mx1 * mx1 + EPSF;
    const float c12 = acc[11][vi] * invN - mx1 * mx2;
    const float c22 = acc[12][vi] * invN - mx2 * mx2 + EPSF;

    const float i00 = c11 * c22 - c12 * c12;
    const float i01 = c02 * c12 - c01 * c22;
    const float i02 = c01 * c12 - c02 * c11;
    const float i11 = c00 * c22 - c02 * c02;
    const float i12 = c01 * c02 - c00 * c12;
    const float i22 = c00 * c11 - c01 * c01;
    const float det = c00 * i00 + c01 * i01 + c02 * i02;
    const float rd  = 1.0f / det;

    const float a0 = (cy0 * i00 + cy1 * i01 + cy2 * i02) * rd;
    const float a1 = (cy0 * i01 + cy1 * i11 + cy2 * i12) * rd;
    const float a2 = (cy0 * i02 + cy1 * i12 + cy2 * i22) * rd;
    const float bb = my - (a0 * mx0 + a1 * mx1 + a2 * mx2);

    const size_t pix = (size_t)h * IMW + w;
    ab[((size_t)n * 4 + 0) * HW + pix] = a0;
    ab[((size_t)n * 4 + 1) * HW + pix] = a1;
    ab[((size_t)n * 4 + 2) * HW + pix] = a2;
    ab[((size_t)n * 4 + 3) * HW + pix] = bb;
  }
}

// ---------------- Stage 2: box(A), box(b), fuse output ----------------
__global__ __launch_bounds__(32)
void gf_stage2(const float* __restrict__ x, const float* __restrict__ ab,
               float* __restrict__ out) {
  __shared__ float raw[4][HTW * HTW];
  __shared__ float vs[4][TSZ * HTW];

  const int n    = blockIdx.z;
  const int h0   = blockIdx.y * TSZ;
  const int w0   = blockIdx.x * TSZ;
  const int lane = threadIdx.x;

  const int r0 = max(h0 - RR, 0);
  const int r1 = min(h0 + TSZ - 1 + RR, IMH - 1);
  const int c0 = max(w0 - RR, 0);
  const int c1 = min(w0 + TSZ - 1 + RR, IMW - 1);
  const int Hc = r1 - r0 + 1;
  const int Wc = c1 - c0 + 1;

  const size_t HW = (size_t)IMH * IMW;

  {
    for (int c = 0; c < 4; ++c) {
      const float* pp = ab + ((size_t)n * 4 + c) * HW + (size_t)r0 * IMW + c0;
      tdm_load_tile_f32(lds_byte_off(&raw[c][0]), pp, Wc, Hc, IMW);
    }
    __builtin_amdgcn_s_wait_tensorcnt(0);
  }
  __syncthreads();

  for (int idx = lane; idx < TSZ * HTW; idx += 32) {
    const int orow = idx / HTW;
    const int cc   = idx - orow * HTW;
    float s0 = 0, s1 = 0, s2 = 0, s3 = 0;
    if (cc < Wc) {
      const int h   = h0 + orow;
      const int rlo = max(h - RR, 0) - r0;
      const int rhi = min(h + RR, IMH - 1) - r0;
      for (int rr = rlo; rr <= rhi; ++rr) {
        const int off = rr * Wc + cc;
        s0 += raw[0][off]; s1 += raw[1][off]; s2 += raw[2][off]; s3 += raw[3][off];
      }
    }
    vs[0][idx] = s0; vs[1][idx] = s1; vs[2][idx] = s2; vs[3][idx] = s3;
  }
  __syncthreads();

  const int d0    = w0 - c0;
  const int Nn    = lane & 15;
  const int khalf = (lane >> 4) * 2;

  v8f acc[4];
  #pragma unroll
  for (int chn = 0; chn < 4; ++chn) acc[chn] = (v8f){0, 0, 0, 0, 0, 0, 0, 0};

  #pragma unroll
  for (int kb = 0; kb < 6; ++kb) {
    const int k0 = kb * 4 + khalf;
    const v2f bfrag = { (iabs(k0     - Nn - d0) <= RR) ? 1.0f : 0.0f,
                        (iabs(k0 + 1 - Nn - d0) <= RR) ? 1.0f : 0.0f };
    #pragma unroll
    for (int chn = 0; chn < 4; ++chn) {
      const v2f afrag = { vs[chn][Nn * HTW + k0], vs[chn][Nn * HTW + k0 + 1] };
      acc[chn] = __builtin_amdgcn_wmma_f32_16x16x4_f32(
          false, afrag, false, bfrag, (short)0, acc[chn], false, false);
    }
  }

  const int hi = lane >> 4;
  const int w  = w0 + Nn;
  const int nhc = min(w + RR, IMW - 1) - max(w - RR, 0) + 1;

  #pragma unroll
  for (int vi = 0; vi < 8; ++vi) {
    const int M = vi + 8 * hi;
    const int h = h0 + M;
    const int nvc = min(h + RR, IMH - 1) - max(h - RR, 0) + 1;
    const float invN = 1.0f / (float)(nvc * nhc);

    const size_t pix = (size_t)h * IMW + w;
    const float x0 = x[((size_t)n * 3 + 0) * HW + pix];
    const float x1 = x[((size_t)n * 3 + 1) * HW + pix];
    const float x2 = x[((size_t)n * 3 + 2) * HW + pix];

    out[(size_t)n * HW + pix] =
        (acc[0][vi] * invN) * x0 + (acc[1][vi] * invN) * x1 +
        (acc[2][vi] * invN) * x2 + (acc[3][vi] * invN);
  }
}

extern "C" void kernel_launch(void* const* d_in, const int* in_sizes, int n_in,
                              void* d_out, int out_size, void* d_ws, size_t ws_size,
                              hipStream_t stream) {
  const float* y = (const float*)d_in[0];   // (4,1,512,512)
  const float* x = (const float*)d_in[1];   // (4,3,512,512)
  float* out = (float*)d_out;               // (4,1,512,512)
  float* ab  = (float*)d_ws;                // 4 planes/batch: A0,A1,A2,b (16.8 MB)

  dim3 grid(IMW / TSZ, IMH / TSZ, NBATCH);
  gf_stage1<<<grid, 32, 0, stream>>>(y, x, ab);
  gf_stage2<<<grid, 32, 0, stream>>>(x, ab, out);
}